// PostProcess_28080496181684
// MI455X (gfx1250) — compile-verified
//
#include <hip/hip_runtime.h>
#include <stdint.h>

#define PP_B    16
#define PP_NC   80
#define PP_K    85
#define PP_A    25200      // 3*80*80 + 3*40*40 + 3*20*20
#define PP_CAP  4096       // candidate cap per image (sorted; top 1024 used)
#define PP_NMSN 1024       // MAX_NMS
#define PP_DET  300
#define PP_BINS 4096       // (bits>>12) bins covering (0.25, 1.0)
#define PP_BIN_BASE 0x3E800u
#define PP_SCORE_TH 0.25f
#define PP_NMS_TH   0.45f
#define PP_MAX_COORD 4096.0f

__device__ __forceinline__ float pp_sigmoid(float x) {
  return 1.0f / (1.0f + __expf(-x));
}

// anchor -> level params; returns pointer to head element base for (b,anchor)
__device__ __forceinline__ const float* pp_head_elem(
    const float* h0, const float* h1, const float* h2,
    int b, int anchor, int* ax, int* ay, int* aa, float* stride) {
  const float* hp; int off, H, W;
  if (anchor < 19200)      { hp = h0; off = 0;     H = 80; W = 80; *stride = 8.0f;  }
  else if (anchor < 24000) { hp = h1; off = 19200; H = 40; W = 40; *stride = 16.0f; }
  else                     { hp = h2; off = 24000; H = 20; W = 20; *stride = 32.0f; }
  int local = anchor - off;
  int hw = H * W;
  int a = local / hw;
  int rem = local - a * hw;
  *aa = a; *ay = rem / W; *ax = rem - (*ay) * W;
  return hp + (size_t)((b * 3 + a) * hw + rem) * PP_K;
}

__device__ __forceinline__ int pp_bin(float s) {
  unsigned bits = __float_as_uint(s);
  int bin = (int)(bits >> 12) - (int)PP_BIN_BASE;
  bin = bin < 0 ? 0 : (bin > PP_BINS - 1 ? PP_BINS - 1 : bin);
  return bin;
}

// ---------------- kernel 1 (fused): decode boxes + objectness + LDS score histogram ----------------
// grid: (ceil(A/256), B), block: 256.  Single pass over the head tensors.
__global__ __launch_bounds__(256) void pp_decode_hist(
    const float* __restrict__ h0, const float* __restrict__ h1,
    const float* __restrict__ h2,
    const float* __restrict__ s0, const float* __restrict__ s1,
    const float* __restrict__ s2,
    float* __restrict__ boxes, float* __restrict__ obj,
    unsigned* __restrict__ hist) {
  __shared__ unsigned lh[PP_BINS];          // 16 KB per-block histogram
  int t = threadIdx.x;
  int b = blockIdx.y;
  int anchor = blockIdx.x * 256 + t;
  for (int i = t; i < PP_BINS; i += 256) lh[i] = 0u;
  __syncthreads();

  if (anchor < PP_A) {
    int x, y, a; float stride;
    const float* e = pp_head_elem(h0, h1, h2, b, anchor, &x, &y, &a, &stride);
    const float* sp = (anchor < 19200) ? s0 : ((anchor < 24000) ? s1 : s2);
    float sx = pp_sigmoid(e[0]);
    float sy = pp_sigmoid(e[1]);
    float sw = pp_sigmoid(e[2]);
    float sh = pp_sigmoid(e[3]);
    float so = pp_sigmoid(e[4]);
    float cx = (sx * 2.0f - 0.5f + (float)x) * stride;
    float cy = (sy * 2.0f - 0.5f + (float)y) * stride;
    float ww = (sw * 2.0f) * (sw * 2.0f) * sp[a * 2 + 0];
    float hh = (sh * 2.0f) * (sh * 2.0f) * sp[a * 2 + 1];
    int gid = b * PP_A + anchor;
    ((float4*)boxes)[gid] = make_float4(cx - ww * 0.5f, cy - hh * 0.5f,
                                        cx + ww * 0.5f, cy + hh * 0.5f);
    obj[gid] = so;
    for (int c = 0; c < PP_NC; ++c) {
      float s = so * pp_sigmoid(e[5 + c]);
      if (s > PP_SCORE_TH) atomicAdd(&lh[pp_bin(s)], 1u);   // ds_add_u32
    }
  }
  __syncthreads();
  for (int i = t; i < PP_BINS; i += 256) {
    unsigned v = lh[i];
    if (v) atomicAdd(&hist[(size_t)b * PP_BINS + i], v);
  }
}

__device__ __forceinline__ float pp_score(const float* h0, const float* h1, const float* h2,
                                          const float* obj, int b, int anchor, int c) {
  int x, y, a; float st;
  const float* e = pp_head_elem(h0, h1, h2, b, anchor, &x, &y, &a, &st);
  return obj[b * PP_A + anchor] * pp_sigmoid(e[5 + c]);
}

// ---------------- kernel 2: find threshold bin (top-down cumulative) ----------------
__global__ __launch_bounds__(1024) void pp_select(const unsigned* __restrict__ hist,
                                                  unsigned* __restrict__ tbinOut) {
  __shared__ unsigned sc[1024];
  __shared__ unsigned carry;
  __shared__ int found;
  __shared__ unsigned tbin;
  int b = blockIdx.x, t = threadIdx.x;
  if (t == 0) { carry = 0u; found = 0; tbin = 0u; }
  __syncthreads();
  for (int c = PP_BINS / 1024 - 1; c >= 0; --c) {
    if (found) break;                       // uniform (barrier-ordered)
    unsigned base = carry;
    unsigned bin = (unsigned)c * 1024u + (unsigned)t;
    unsigned v = hist[(size_t)b * PP_BINS + bin];
    unsigned q = 1023u - (unsigned)t;       // q grows as bin decreases
    sc[q] = v;
    __syncthreads();
    for (int o = 1; o < 1024; o <<= 1) {
      unsigned tv = (q >= (unsigned)o) ? sc[q - o] : 0u;
      __syncthreads();
      sc[q] += tv;
      __syncthreads();
    }
    unsigned incl = base + sc[q];
    if (incl >= (unsigned)PP_NMSN && incl - v < (unsigned)PP_NMSN) { tbin = bin; found = 1; }
    __syncthreads();
    if (t == 0) carry = base + sc[1023];
    __syncthreads();
  }
  if (t == 0) tbinOut[b] = tbin;
}

// ---------------- kernel 3: compact candidates in bins >= T (coalesced class-per-thread) ----------------
__global__ void pp_compact(const float* __restrict__ h0, const float* __restrict__ h1,
                           const float* __restrict__ h2, const float* __restrict__ obj,
                           const unsigned* __restrict__ tbin,
                           unsigned* __restrict__ cnt,
                           unsigned long long* __restrict__ cand) {
  long long gid = (long long)blockIdx.x * blockDim.x + threadIdx.x;
  if (gid >= (long long)PP_B * PP_A * PP_NC) return;
  int b = (int)(gid / ((long long)PP_A * PP_NC));
  int r = (int)(gid - (long long)b * PP_A * PP_NC);
  int anchor = r / PP_NC;
  int c = r - anchor * PP_NC;
  float s = pp_score(h0, h1, h2, obj, b, anchor, c);
  if (s > PP_SCORE_TH && pp_bin(s) >= (int)tbin[b]) {
    unsigned pos = atomicAdd(&cnt[b], 1u);
    if (pos < (unsigned)PP_CAP) {
      unsigned flat = (unsigned)(anchor * PP_NC + c);
      unsigned long long key =
          ((unsigned long long)__float_as_uint(s) << 32) |
          (unsigned long long)(0xFFFFFFFFu - flat);   // score desc, idx asc
      cand[(size_t)b * PP_CAP + pos] = key;
    }
  }
}

// ---------------- kernel 4: bitonic sort, emit top-1024 ----------------
__global__ __launch_bounds__(1024) void pp_sort(const unsigned long long* __restrict__ cand,
                                                const unsigned* __restrict__ cnt,
                                                float* __restrict__ topS,
                                                unsigned* __restrict__ topI) {
  __shared__ unsigned long long k[PP_CAP];   // 32 KB LDS
  int b = blockIdx.x, t = threadIdx.x;
  unsigned n = cnt[b]; if (n > (unsigned)PP_CAP) n = PP_CAP;
  for (int i = t; i < PP_CAP; i += 1024)
    k[i] = (i < (int)n) ? ~cand[(size_t)b * PP_CAP + i] : ~0ULL;  // ascending ~key == descending key
  __syncthreads();
  for (unsigned kk = 2; kk <= (unsigned)PP_CAP; kk <<= 1) {
    for (unsigned j = kk >> 1; j > 0; j >>= 1) {
      for (int i = t; i < PP_CAP; i += 1024) {
        unsigned ixj = (unsigned)i ^ j;
        if (ixj > (unsigned)i) {
          bool up = (((unsigned)i & kk) == 0u);
          unsigned long long A = k[i], Bv = k[ixj];
          if ((A > Bv) == up) { k[i] = Bv; k[ixj] = A; }
        }
      }
      __syncthreads();
    }
  }
  unsigned long long p = ~k[t];              // rank t (descending)
  unsigned bits = (unsigned)(p >> 32);
  unsigned idx = 0xFFFFFFFFu - (unsigned)(p & 0xFFFFFFFFu);
  topS[b * PP_NMSN + t] = bits ? __uint_as_float(bits) : 0.0f;
  topI[b * PP_NMSN + t] = bits ? idx : 0u;
}

// ---------------- kernel 5: NMS + top-300 emit ----------------
__global__ __launch_bounds__(1024) void pp_nms(const float* __restrict__ boxes,
                                               const float* __restrict__ topS,
                                               const unsigned* __restrict__ topI,
                                               float* __restrict__ out) {
  __shared__ float4   sb[PP_NMSN];     // 16 KB
  __shared__ float    sarea[PP_NMSN];  //  4 KB
  __shared__ unsigned skeep[PP_NMSN];  //  4 KB
  __shared__ unsigned sscan[PP_NMSN];  //  4 KB
  int b = blockIdx.x, r = threadIdx.x;

  float s = topS[b * PP_NMSN + r];
  unsigned idx = topI[b * PP_NMSN + r];
  unsigned anchor = idx / PP_NC;
  unsigned cls = idx - anchor * PP_NC;
  if (s <= 0.0f) { anchor = 0; cls = 0; }

  // CDNA5 async global->LDS gather of this lane's box (16B per lane)
  uint64_t ga = (uint64_t)(uintptr_t)(boxes + (size_t)(b * PP_A + anchor) * 4);
  uint32_t la = (uint32_t)(uintptr_t)(void*)&sb[r];   // flat LDS addr[31:0] == LDS offset
  asm volatile("global_load_async_to_lds_b128 %0, %1, off" :: "v"(la), "v"(ga) : "memory");
  asm volatile("s_wait_asynccnt 0" ::: "memory");
  __syncthreads();

  float4 raw = sb[r];
  float off = (float)cls * PP_MAX_COORD;
  float x1 = raw.x + off, y1 = raw.y + off, x2 = raw.z + off, y2 = raw.w + off;
  sb[r] = make_float4(x1, y1, x2, y2);
  float myArea = (x2 - x1) * (y2 - y1);
  sarea[r] = myArea;
  skeep[r] = (s > 0.0f) ? 1u : 0u;
  __syncthreads();

  for (int i = 0; i < PP_NMSN; ++i) {
    if (skeep[i]) {                         // uniform LDS read
      if (r > i && skeep[r]) {
        float4 bi = sb[i];
        float ix1 = fmaxf(x1, bi.x), iy1 = fmaxf(y1, bi.y);
        float ix2 = fminf(x2, bi.z), iy2 = fminf(y2, bi.w);
        float iw = fmaxf(ix2 - ix1, 0.0f), ih = fmaxf(iy2 - iy1, 0.0f);
        float inter = iw * ih;
        float iou = inter / (myArea + sarea[i] - inter + 1e-7f);
        if (iou > PP_NMS_TH) skeep[r] = 0u;
      }
    }
    __syncthreads();
  }

  // rank kept entries (exclusive scan over keep flags)
  unsigned flag = skeep[r];
  sscan[r] = flag;
  __syncthreads();
  for (int o = 1; o < PP_NMSN; o <<= 1) {
    unsigned tv = (r >= o) ? sscan[r - o] : 0u;
    __syncthreads();
    sscan[r] += tv;
    __syncthreads();
  }
  unsigned rank = sscan[r] - flag;
  unsigned total = sscan[PP_NMSN - 1];

  float* oB = out;
  float* oS = out + (size_t)PP_B * PP_DET * 4;
  float* oL = out + (size_t)PP_B * PP_DET * 5;
  if (flag && rank < (unsigned)PP_DET) {
    size_t ob = ((size_t)b * PP_DET + rank) * 4;
    oB[ob + 0] = raw.x; oB[ob + 1] = raw.y; oB[ob + 2] = raw.z; oB[ob + 3] = raw.w;
    oS[b * PP_DET + rank] = s;
    oL[b * PP_DET + rank] = (float)cls;
  }
  if (r < PP_DET && (unsigned)r >= total) {
    size_t ob = ((size_t)b * PP_DET + r) * 4;
    oB[ob + 0] = 0.0f; oB[ob + 1] = 0.0f; oB[ob + 2] = 0.0f; oB[ob + 3] = 0.0f;
    oS[b * PP_DET + r] = 0.0f;
    oL[b * PP_DET + r] = -1.0f;
  }
}

extern "C" void kernel_launch(void* const* d_in, const int* in_sizes, int n_in,
                              void* d_out, int out_size, void* d_ws, size_t ws_size,
                              hipStream_t stream) {
  (void)in_sizes; (void)n_in; (void)out_size; (void)ws_size;
  const float* h0 = (const float*)d_in[0];
  const float* s0 = (const float*)d_in[2];
  const float* h1 = (const float*)d_in[3];
  const float* s1 = (const float*)d_in[5];
  const float* h2 = (const float*)d_in[6];
  const float* s2 = (const float*)d_in[8];

  char* ws = (char*)d_ws;
  size_t o = 0;
  float* boxes = (float*)(ws + o);            o += (size_t)PP_B * PP_A * 4 * sizeof(float);   // 6.45 MB
  float* obj   = (float*)(ws + o);            o += (size_t)PP_B * PP_A * sizeof(float);       // 1.61 MB
  size_t zero_off = o;
  unsigned* hist = (unsigned*)(ws + o);       o += (size_t)PP_B * PP_BINS * sizeof(unsigned); // 256 KB
  unsigned* cnt  = (unsigned*)(ws + o);       o += 64;
  size_t zero_len = o - zero_off;
  unsigned* tbin = (unsigned*)(ws + o);       o += 64;
  unsigned long long* cand = (unsigned long long*)(ws + o); o += (size_t)PP_B * PP_CAP * 8;   // 512 KB
  float*    topS = (float*)(ws + o);          o += (size_t)PP_B * PP_NMSN * sizeof(float);
  unsigned* topI = (unsigned*)(ws + o);       o += (size_t)PP_B * PP_NMSN * sizeof(unsigned);

  hipMemsetAsync(ws + zero_off, 0, zero_len, stream);

  {
    dim3 grid((PP_A + 255) / 256, PP_B);
    pp_decode_hist<<<grid, 256, 0, stream>>>(h0, h1, h2, s0, s1, s2, boxes, obj, hist);
  }
  pp_select<<<PP_B, 1024, 0, stream>>>(hist, tbin);
  {
    long long n = (long long)PP_B * PP_A * PP_NC;
    int blocks = (int)((n + 255) / 256);
    pp_compact<<<blocks, 256, 0, stream>>>(h0, h1, h2, obj, tbin, cnt, cand);
  }
  pp_sort<<<PP_B, 1024, 0, stream>>>(cand, cnt, topS, topI);
  pp_nms<<<PP_B, 1024, 0, stream>>>(boxes, topS, topI, (float*)d_out);
}